// RtrlRNNCellFwd_67053029425247
// MI455X (gfx1250) — compile-verified
//
#include <hip/hip_runtime.h>

// RTRL RNN cell: B=32, H=I=128.
// Heavy math: per-batch GEMM  new_S = M^T (128x128) x S (128 x ncols) + diag,
// where M[b][n,k] = (1 - tanh(prev_s[b,n])^2) * R[n,k].
// fp32 WMMA (v_wmma_f32_16x16x4_f32): the problem runs at ~32 FLOP/byte fp32,
// right at the MI455X fp32 roofline knee (23.3 TB/s HBM), so fp32 matrix ops
// are the correct precision/path.
//
// M is stored PAIR-INTERLEAVED: Mp[b][n>>1][k][n&1], so a WMMA A-fragment
// (two consecutive K rows for one output column) is a single aligned
// ds_load_b64 into an even VGPR pair -- no repacking moves.
// NCOLS is a template constant so every global load/store/prefetch in the
// fully-unrolled K-loop uses immediate offsets from one base address.

typedef __attribute__((ext_vector_type(2))) float v2f;
typedef __attribute__((ext_vector_type(8))) float v8f;

#define Bsz 32
#define Hsz 128
#define Isz 128

// ---------------------------------------------------------------------------
// Kernel 1: dense RNN step + build pair-interleaved M[b], stash prev_h.
// One block per batch, 128 threads.
// ---------------------------------------------------------------------------
__global__ void rtrl_small_kernel(const float* __restrict__ prev_s,
                                  const float* __restrict__ x,
                                  const float* __restrict__ Wi,
                                  const float* __restrict__ bi,
                                  const float* __restrict__ R,
                                  float* __restrict__ out_curr_s,
                                  float* __restrict__ out_curr_out,
                                  float* __restrict__ Mws,
                                  float* __restrict__ ph_ws) {
    __shared__ float sh_ph[Hsz];
    __shared__ float sh_ds[Hsz];
    __shared__ float sh_x[Isz];

    const int b = blockIdx.x;
    const int t = threadIdx.x;

    const float ph = tanhf(prev_s[b * Hsz + t]);
    sh_ph[t] = ph;
    sh_ds[t] = 1.0f - ph * ph;
    sh_x[t]  = x[b * Isz + t];
    ph_ws[b * Hsz + t] = ph;
    __syncthreads();

    // curr_s[b,t] = bi[t] + sum_i x[b,i]*Wi[i,t] + sum_n prev_h[b,n]*R[n,t]
    float acc = bi[t];
    #pragma unroll 4
    for (int i = 0; i < Isz; ++i) acc += sh_x[i] * Wi[i * Hsz + t];
    #pragma unroll 4
    for (int n = 0; n < Hsz; ++n) acc += sh_ph[n] * R[n * Hsz + t];

    out_curr_s[b * Hsz + t]   = acc;
    out_curr_out[b * Hsz + t] = tanhf(acc);

    // Pair-interleaved M: Mp[(n>>1)*256 + t*2 + (n&1)] = d_s[n] * R[n,t]
    float* Mb = Mws + (size_t)b * Hsz * Hsz;
    #pragma unroll 4
    for (int n = 0; n < Hsz; ++n)
        Mb[(n >> 1) * (2 * Hsz) + t * 2 + (n & 1)] = sh_ds[n] * R[n * Hsz + t];
}

// ---------------------------------------------------------------------------
// Kernel 2: per-batch WMMA GEMM  D = M^T x S  (+ diagonal injection).
// Grid: (column strip groups, batch). Block = 256 threads = 8 waves.
// Each wave produces a 128-row x 16-column output strip:
//   8 k-tiles x 32 fully-unrolled K-steps of v_wmma_f32_16x16x4_f32.
// M[b] (64 KB, pair-interleaved) staged in LDS, reused by all 8 waves.
// diagvec == nullptr means inject 1.0 on the diagonal (S_B case).
// ---------------------------------------------------------------------------
template <int NCOLS>
__global__ void __launch_bounds__(256)
rtrl_wmma_gemm(const float* __restrict__ S,
               float* __restrict__ D,
               const float* __restrict__ Mws,
               const float* __restrict__ diagvec) {
    __shared__ float Msh[Hsz * Hsz];   // 64 KB, pair-interleaved M[b]

    const int b   = blockIdx.y;
    const int tid = threadIdx.x;

    // Cooperative 128-bit staging of packed M[b] into LDS (straight copy).
    {
        const float4* Mg4 = (const float4*)(Mws + (size_t)b * Hsz * Hsz);
        float4*       Ms4 = (float4*)Msh;
        #pragma unroll
        for (int idx = tid; idx < (Hsz * Hsz) / 4; idx += 256)
            Ms4[idx] = Mg4[idx];
    }
    __syncthreads();

    const int wave = tid >> 5;
    const int lane = tid & 31;
    const int cset = lane & 15;        // column within 16-wide strip
    const int rsel = lane >> 4;        // 0: lanes 0-15, 1: lanes 16-31

    const size_t c0 = ((size_t)blockIdx.x * 8 + wave) * 16;
    const size_t c  = c0 + cset;       // this lane's global column

    const float* src = S + (size_t)b * Hsz * NCOLS;
    float*       dst = D + (size_t)b * Hsz * NCOLS;

    // Diagonal injection data (column is fixed per lane).
    const int j = (int)(c & (size_t)(Hsz - 1));   // c % H  -> last-dim index
    const int i = (int)(c >> 7);                  // c / H  -> middle index
    const float addval = diagvec ? diagvec[b * Isz + i] : 1.0f;

    // Single per-lane base; all K-step deltas are compile-time immediates
    // (global: <= 8.32 MB < 24-bit signed; LDS: <= 64 KB < 16-bit DS offset).
    const float* bbase = src + (size_t)rsel * NCOLS + c;   // row n0+rsel, col c
    const float* abase = Msh + rsel * (2 * Hsz) + cset * 2;

    v8f acc[8] = {};

    #pragma unroll
    for (int n0 = 0; n0 < Hsz; n0 += 4) {
        // B tile 4x16 fp32 (2 VGPRs):
        //   VGPR0: lanes 0-15 -> row n0,   lanes 16-31 -> row n0+1
        //   VGPR1: lanes 0-15 -> row n0+2, lanes 16-31 -> row n0+3
        v2f bt;
        bt.x = bbase[(size_t)n0 * NCOLS];
        bt.y = bbase[(size_t)(n0 + 2) * NCOLS];

        // Latency prefetch 16 rows (4 K-steps) ahead; tensors stream once,
        // so this is purely a latency hint into GL2.
        if (n0 + 16 < Hsz)
            __builtin_prefetch(&bbase[(size_t)(n0 + 16) * NCOLS], 0, 1);

        // A tile 16x4 fp32: one aligned ds_load_b64 per k-tile thanks to the
        // pair-interleaved layout (rows n0+2*rsel and n0+2*rsel+1 adjacent).
        const int aoff = (n0 >> 1) * (2 * Hsz);

        #pragma unroll
        for (int kt = 0; kt < 8; ++kt) {
            const v2f at = *(const v2f*)(abase + aoff + kt * 32);
            acc[kt] = __builtin_amdgcn_wmma_f32_16x16x4_f32(
                /*neg_a=*/false, at, /*neg_b=*/false, bt,
                /*c_mod=*/(short)0, acc[kt],
                /*reuse_a=*/false, /*reuse_b=*/false);
        }
    }

    // Epilogue: diagonal injection + store.
    // C/D layout: VGPR v -> row (kt*16 + v + 8*rsel), column fixed = c.
    float* dbase = dst + c;
    #pragma unroll
    for (int kt = 0; kt < 8; ++kt) {
        #pragma unroll
        for (int v = 0; v < 8; ++v) {
            const int k = kt * 16 + v + 8 * rsel;
            float val = acc[kt][v];
            if (k == j) val += addval;
            dbase[(size_t)k * NCOLS] = val;
        }
    }
}

// ---------------------------------------------------------------------------
// Host launcher
// ---------------------------------------------------------------------------
extern "C" void kernel_launch(void* const* d_in, const int* in_sizes, int n_in,
                              void* d_out, int out_size, void* d_ws, size_t ws_size,
                              hipStream_t stream) {
    (void)in_sizes; (void)n_in; (void)out_size; (void)ws_size;

    const float* prev_s = (const float*)d_in[0];
    const float* S_R    = (const float*)d_in[1];
    const float* S_W    = (const float*)d_in[2];
    const float* S_B    = (const float*)d_in[3];
    const float* x      = (const float*)d_in[4];
    const float* Wi     = (const float*)d_in[5];
    const float* bi     = (const float*)d_in[6];
    const float* R      = (const float*)d_in[7];

    float* out = (float*)d_out;
    const size_t SMALL = (size_t)Bsz * Hsz;                  // 4096
    const size_t SBIG  = (size_t)Bsz * Hsz * Isz * Hsz;      // 67108864
    const size_t SMED  = (size_t)Bsz * Hsz * Hsz;            // 524288

    float* out_curr_s = out;
    float* out_SR     = out_curr_s + SMALL;
    float* out_SW     = out_SR + SBIG;
    float* out_SB     = out_SW + SBIG;
    float* out_co     = out_SB + SMED;

    // Workspace: pair-interleaved M (2 MB) then prev_h (16 KB).
    float* Mws   = (float*)d_ws;
    float* ph_ws = Mws + (size_t)Bsz * Hsz * Hsz;

    rtrl_small_kernel<<<dim3(Bsz), dim3(Hsz), 0, stream>>>(
        prev_s, x, Wi, bi, R, out_curr_s, out_co, Mws, ph_ws);

    constexpr int NC_BIG = Isz * Hsz;       // 16384
    dim3 gridBig(NC_BIG / 128, Bsz);        // 8 waves x 16 cols = 128 cols/block
    dim3 gridSB(1, Bsz);                    // 128 cols = one block
    dim3 blk(256);

    // new_S_R = M^T x S_R + diag(prev_h)
    rtrl_wmma_gemm<NC_BIG><<<gridBig, blk, 0, stream>>>(S_R, out_SR, Mws, ph_ws);
    // new_S_W = M^T x S_W + diag(x)
    rtrl_wmma_gemm<NC_BIG><<<gridBig, blk, 0, stream>>>(S_W, out_SW, Mws, x);
    // new_S_B = M^T x S_B + I
    rtrl_wmma_gemm<Hsz><<<gridSB, blk, 0, stream>>>(S_B, out_SB, Mws, nullptr);
}